// StabilityGNN_51857435132135
// MI455X (gfx1250) — compile-verified
//
#include <hip/hip_runtime.h>
#include <math.h>

typedef __attribute__((ext_vector_type(2))) float v2f;
typedef __attribute__((ext_vector_type(8))) float v8f;

#define N_NODES 8192
#define F_IN    256
#define H_DIM   128
#define RES2    400
#define C_OUT   10
#define NEG_SLOPE 0.2f

__device__ __forceinline__ float lrelu(float x) { return x > 0.f ? x : NEG_SLOPE * x; }

// ---------------------------------------------------------------------------
// Kernel 0: h = X @ W^T   (8192x256 * 256x128) via V_WMMA_F32_16X16X4_F32
// One 256-thread block per 16-row band; wave w owns the 16-col tile [16w,16w+16).
// A-tile (16x4 f32, 2 VGPRs): lane l -> M=l&15, K=(l>>4)*2 + {0,1}
// B-tile (4x16 f32, 2 VGPRs): lane l -> N=l&15, K=(l>>4)*2 + {0,1};  B[k][n]=W[n][k]
// ---------------------------------------------------------------------------
__global__ __launch_bounds__(256) void k_hgemm(const float* __restrict__ X,
                                               const float* __restrict__ W,
                                               float* __restrict__ h) {
  const int lane = threadIdx.x & 31;
  const int wave = threadIdx.x >> 5;     // 0..7
  const int lm   = lane & 15;
  const int lh   = lane >> 4;            // 0 or 1
  const int rowbase = blockIdx.x * 16;
  const int colbase = wave * 16;

  const float* arow = X + (size_t)(rowbase + lm) * F_IN + lh * 2;
  const float* brow = W + (size_t)(colbase + lm) * F_IN + lh * 2;

  v8f acc = {};
  for (int k = 0; k < F_IN; k += 4) {
    v2f a, b;
    a.x = arow[k];  a.y = arow[k + 1];
    b.x = brow[k];  b.y = brow[k + 1];
    acc = __builtin_amdgcn_wmma_f32_16x16x4_f32(false, a, false, b,
                                                (short)0, acc, false, false);
  }
  // C/D layout: VGPR v -> M = v + lh*8, N = lm
#pragma unroll
  for (int v = 0; v < 8; ++v)
    h[(size_t)(rowbase + v + lh * 8) * H_DIM + colbase + lm] = acc[v];
}

// ---------------------------------------------------------------------------
// Kernel 1: s_i[n] = h[n].a_i, s_j[n] = h[n].a_j ; c = a_t.topo ; zero colsum
// ---------------------------------------------------------------------------
__global__ __launch_bounds__(256) void k_scores(const float* __restrict__ h,
                                                const float* __restrict__ att,
                                                const float* __restrict__ topo,
                                                float* __restrict__ s_i,
                                                float* __restrict__ s_j,
                                                float* __restrict__ cptr,
                                                float* __restrict__ colsum) {
  const int n = blockIdx.x * 256 + threadIdx.x;
  const float* hr = h + (size_t)n * H_DIM;
  float si = 0.f, sj = 0.f;
  for (int f = 0; f < H_DIM; ++f) {
    float v = hr[f];
    si += v * att[f];
    sj += v * att[H_DIM + f];
  }
  s_i[n] = si;
  s_j[n] = sj;
  if (blockIdx.x == 0) {
    if (threadIdx.x < H_DIM) colsum[threadIdx.x] = 0.f;
    if (threadIdx.x == 0) {
      float cacc = 0.f;
      for (int t = 0; t < RES2; ++t) cacc += att[2 * H_DIM + t] * topo[t];
      *cptr = cacc;
    }
  }
}

// ---------------------------------------------------------------------------
// Kernel 2: masked softmax attention + aggregation.
// One block per 16 rows. adj rows read from HBM exactly once -> LDS bitmask.
// Pass1: bitmask + row max + degree (ballot_w32, wave32 shuffles)
// Pass2: row sum of exp (incl. (N-deg)*exp(-M) for masked-out zeros)
// Pass3: stage 16x64 weight chunks in LDS, V_WMMA_F32_16X16X4_F32 against h.
// Epilogue: ELU + per-column sum -> atomicAdd into colsum[128].
// ---------------------------------------------------------------------------
__global__ __launch_bounds__(256) void k_attagg(const int* __restrict__ adj,
                                                const float* __restrict__ h,
                                                const float* __restrict__ s_i,
                                                const float* __restrict__ s_j,
                                                const float* __restrict__ cptr,
                                                float* __restrict__ colsum) {
  __shared__ unsigned lmask[16][256];    // 16 KB : 16 rows x 8192 bits
  __shared__ float    lsj[N_NODES];      // 32 KB : staged s_j
  __shared__ float    lw[16 * 64];       //  4 KB : weight chunk
  __shared__ float    rowM[16], rowInvS[16], rowSi[16];
  __shared__ int      rowCnt[16];

  const int tid  = threadIdx.x;
  const int lane = tid & 31;
  const int wave = tid >> 5;             // 0..7
  const int row0 = blockIdx.x * 16;
  const float c  = *cptr;

  for (int i = tid; i < N_NODES; i += 256) lsj[i] = s_j[i];
  if (tid < 16) rowSi[tid] = s_i[row0 + tid];
  __syncthreads();

  // ---- Pass 1: each wave owns rows 2*wave, 2*wave+1 ----
  for (int rr = 0; rr < 2; ++rr) {
    const int r = wave * 2 + rr;
    const int R = row0 + r;
    const float siR = rowSi[r];
    const int* arow = adj + (size_t)R * N_NODES;
    float m = -3.0e38f;
    int cnt = 0;
    for (int i = 0; i < 256; ++i) {
      const int j = i * 32 + lane;
      const bool bit = (arow[j] != 0) && (j != R);
      const unsigned mw = __builtin_amdgcn_ballot_w32(bit);
      if (lane == 0) lmask[r][i] = mw;
      if (bit) {
        m = fmaxf(m, lrelu(siR + lsj[j] + c));
        cnt++;
      }
    }
    for (int off = 16; off; off >>= 1) {
      m    = fmaxf(m, __shfl_xor(m, off, 32));
      cnt += __shfl_xor(cnt, off, 32);
    }
    if (lane == 0) {
      rowM[r]   = (cnt > 0) ? fmaxf(0.f, m) : 0.f;  // unmasked scores are 0
      rowCnt[r] = cnt;
    }
  }
  __syncthreads();

  // ---- Pass 2: softmax denominators ----
  for (int rr = 0; rr < 2; ++rr) {
    const int r = wave * 2 + rr;
    const float siR = rowSi[r];
    const float M   = rowM[r];
    float s = 0.f;
    for (int i = 0; i < 256; ++i) {
      const unsigned mw = lmask[r][i];
      if ((mw >> lane) & 1u) {
        const int j = i * 32 + lane;
        s += __expf(lrelu(siR + lsj[j] + c) - M);
      }
    }
    for (int off = 16; off; off >>= 1) s += __shfl_xor(s, off, 32);
    if (lane == 0) {
      const float S = s + (float)(N_NODES - rowCnt[r]) * __expf(-M);
      rowInvS[r] = 1.f / S;
    }
  }
  __syncthreads();

  // ---- Pass 3: WMMA aggregation  agg(16x128) = Wgt(16x8192) * h(8192x128) ----
  const int lm = lane & 15;
  const int lh = lane >> 4;
  const int colbase = wave * 16;
  const int r_stage = tid & 15;          // row this thread stages weights for
  const int kslot   = tid >> 4;          // 0..15, covers 4 k's each
  const float siS = rowSi[r_stage];
  const float MS  = rowM[r_stage];
  const float ivS = rowInvS[r_stage];

  v8f acc = {};
  for (int kbase = 0; kbase < N_NODES; kbase += 64) {
#pragma unroll
    for (int kk = 0; kk < 4; ++kk) {
      const int j = kbase + kslot * 4 + kk;
      const unsigned bit = (lmask[r_stage][j >> 5] >> (j & 31)) & 1u;
      float wgt = 0.f;
      if (bit) wgt = __expf(lrelu(siS + lsj[j] + c) - MS) * ivS;
      lw[r_stage * 64 + kslot * 4 + kk] = wgt;
    }
    __syncthreads();

    const float* hb = h + (size_t)(kbase + lh * 2) * H_DIM + colbase + lm;
#pragma unroll 4
    for (int step = 0; step < 16; ++step) {
      v2f a, b;
      const int kl = step * 4 + lh * 2;
      a.x = lw[lm * 64 + kl];
      a.y = lw[lm * 64 + kl + 1];
      b.x = hb[(step * 4)     * H_DIM];
      b.y = hb[(step * 4 + 1) * H_DIM];
      acc = __builtin_amdgcn_wmma_f32_16x16x4_f32(false, a, false, b,
                                                  (short)0, acc, false, false);
    }
    __syncthreads();
  }

  // ---- Epilogue: ELU + column reduction over this 16-row block ----
  float partial = 0.f;
#pragma unroll
  for (int v = 0; v < 8; ++v) {
    const float x = acc[v];
    partial += (x > 0.f) ? x : (__expf(x) - 1.f);
  }
  partial += __shfl_xor(partial, 16, 32);   // lane l & l+16 share a column
  if (lane < 16) atomicAdd(&colsum[colbase + lane], partial);
}

// ---------------------------------------------------------------------------
// Kernel 3: logits = (colsum/N) @ out_W^T + out_b  (10 outputs)
// ---------------------------------------------------------------------------
__global__ void k_final(const float* __restrict__ colsum,
                        const float* __restrict__ outW,
                        const float* __restrict__ outB,
                        float* __restrict__ logits) {
  const int cidx = threadIdx.x;
  if (cidx < C_OUT) {
    float acc = outB[cidx];
    for (int f = 0; f < H_DIM; ++f)
      acc += (colsum[f] * (1.f / (float)N_NODES)) * outW[cidx * H_DIM + f];
    logits[cidx] = acc;
  }
}

// ---------------------------------------------------------------------------
extern "C" void kernel_launch(void* const* d_in, const int* in_sizes, int n_in,
                              void* d_out, int out_size, void* d_ws, size_t ws_size,
                              hipStream_t stream) {
  (void)in_sizes; (void)n_in; (void)out_size; (void)ws_size;
  const float* X    = (const float*)d_in[0];   // node_features 8192x256
  const int*   adj  = (const int*)  d_in[1];   // adj_matrix    8192x8192
  const float* topo = (const float*)d_in[2];   // topo_emb      400
  const float* W    = (const float*)d_in[3];   // W             128x256
  const float* att  = (const float*)d_in[4];   // att_vec       656
  const float* outW = (const float*)d_in[5];   // out_W         10x128
  const float* outB = (const float*)d_in[6];   // out_b         10
  float* logits = (float*)d_out;

  float* ws     = (float*)d_ws;
  float* h      = ws;                                   // 8192*128
  float* s_i    = h + (size_t)N_NODES * H_DIM;          // 8192
  float* s_j    = s_i + N_NODES;                        // 8192
  float* cptr   = s_j + N_NODES;                        // 1 (+pad)
  float* colsum = cptr + 4;                             // 128

  k_hgemm <<<N_NODES / 16,  256, 0, stream>>>(X, W, h);
  k_scores<<<N_NODES / 256, 256, 0, stream>>>(h, att, topo, s_i, s_j, cptr, colsum);
  k_attagg<<<N_NODES / 16,  256, 0, stream>>>(adj, h, s_i, s_j, cptr, colsum);
  k_final <<<1, 32, 0, stream>>>(colsum, outW, outB, logits);
}